// UserGNNEncoder_11768210391489
// MI455X (gfx1250) — compile-verified
//
#include <hip/hip_runtime.h>

typedef float v2f __attribute__((ext_vector_type(2)));
typedef float v8f __attribute__((ext_vector_type(8)));

__device__ __forceinline__ v8f wmma4(v2f a, v2f b, v8f c) {
  // D = A(16x4 f32) * B(4x16 f32) + C(16x16 f32)
  return __builtin_amdgcn_wmma_f32_16x16x4_f32(false, a, false, b, (short)0, c,
                                               false, false);
}

__global__ void zero_f32(float* __restrict__ p, long n) {
  long i = (long)blockIdx.x * blockDim.x + threadIdx.x;
  long st = (long)gridDim.x * blockDim.x;
  for (; i < n; i += st) p[i] = 0.0f;
}

__global__ void count_edges(const int* __restrict__ dst, float* __restrict__ cnt,
                            int E) {
  int i = blockIdx.x * blockDim.x + threadIdx.x;
  if (i < E) atomicAdd(&cnt[dst[i]], 1.0f);
}

// One edge per block: coalesced feature read + coalesced float atomics.
__global__ void scatter_add(const float* __restrict__ xsrc,
                            const int* __restrict__ src,
                            const int* __restrict__ dst,
                            float* __restrict__ agg, int D) {
  int e = blockIdx.x;
  int s = src[e], d = dst[e];
  const float* xp = xsrc + (size_t)s * D;
  float* ap = agg + (size_t)d * D;
  for (int c = threadIdx.x; c < D; c += blockDim.x) atomicAdd(&ap[c], xp[c]);
}

// out[m,n] = act( (agg[m,:]/max(cnt[m],1)) @ wl + xdst[m,:] @ wr + bias[n] )
// Block = NOUT threads (NOUT/32 waves); each wave owns 32 output columns
// (two 16x16 WMMA accumulators). Block covers 16 rows; K via 16x16x4 f32 WMMA.
template <int K1, int K2, int NOUT, bool RELU, bool SCALE>
__global__ void __launch_bounds__(NOUT)
sage_gemm(const float* __restrict__ agg, const float* __restrict__ cnt,
          const float* __restrict__ xdst, const float* __restrict__ wl,
          const float* __restrict__ wr, const float* __restrict__ bias,
          float* __restrict__ out) {
  __shared__ float smA[16 * K1];
  __shared__ float smX[16 * (K2 > 0 ? K2 : 1)];
  __shared__ float rs[16];

  const int row0 = blockIdx.x * 16;

  if (SCALE) {
    if (threadIdx.x < 16) {
      float c = cnt[row0 + threadIdx.x];
      rs[threadIdx.x] = 1.0f / fmaxf(c, 1.0f);
    }
    __syncthreads();
  }

  {  // stage mean tile (row-scaled aggregate) into LDS, float4 coalesced
    const int V = K1 / 4;
    for (int i = threadIdx.x; i < 16 * V; i += NOUT) {
      int r = i / V, c = i % V;
      float4 v = ((const float4*)(agg + (size_t)(row0 + r) * K1))[c];
      if (SCALE) {
        float s = rs[r];
        v.x *= s; v.y *= s; v.z *= s; v.w *= s;
      }
      ((float4*)(smA + r * K1))[c] = v;
    }
  }
  if (K2 > 0) {  // stage x_dst tile
    const int V = K2 / 4;
    for (int i = threadIdx.x; i < 16 * V; i += NOUT) {
      int r = i / V, c = i % V;
      ((float4*)(smX + r * K2))[c] =
          ((const float4*)(xdst + (size_t)(row0 + r) * K2))[c];
    }
  }
  __syncthreads();

  const int lane = threadIdx.x & 31;
  const int wave = threadIdx.x >> 5;
  const int lo = lane & 15;
  const int hi = lane >> 4;
  const int colA = wave * 32 + lo;
  const int colB = colA + 16;

  v8f acc0 = {0.f, 0.f, 0.f, 0.f, 0.f, 0.f, 0.f, 0.f};
  v8f acc1 = {0.f, 0.f, 0.f, 0.f, 0.f, 0.f, 0.f, 0.f};

#pragma unroll 4
  for (int k0 = 0; k0 < K1; k0 += 4) {
    const int kk = k0 + 2 * hi;  // A vgpr j holds A[m=lo, k=kk+j]
    v2f a;
    a.x = smA[lo * K1 + kk];
    a.y = smA[lo * K1 + kk + 1];
    const float* w0 = wl + (size_t)kk * NOUT;  // B vgpr j holds B[kk+j, col]
    v2f b0, b1;
    b0.x = w0[colA];        b1.x = w0[colB];
    b0.y = w0[NOUT + colA]; b1.y = w0[NOUT + colB];
    acc0 = wmma4(a, b0, acc0);
    acc1 = wmma4(a, b1, acc1);
  }
#pragma unroll 4
  for (int k0 = 0; k0 < K2; k0 += 4) {
    const int kk = k0 + 2 * hi;
    v2f a;
    a.x = smX[lo * K2 + kk];
    a.y = smX[lo * K2 + kk + 1];
    const float* w0 = wr + (size_t)kk * NOUT;
    v2f b0, b1;
    b0.x = w0[colA];        b1.x = w0[colB];
    b0.y = w0[NOUT + colA]; b1.y = w0[NOUT + colB];
    acc0 = wmma4(a, b0, acc0);
    acc1 = wmma4(a, b1, acc1);
  }

  const float ba = bias[colA];
  const float bb = bias[colB];
#pragma unroll
  for (int j = 0; j < 8; ++j) {  // C vgpr j = C[m=j+8*hi, n=lo]
    int r = row0 + j + 8 * hi;
    float v0 = acc0[j] + ba;
    float v1 = acc1[j] + bb;
    if (RELU) { v0 = fmaxf(v0, 0.f); v1 = fmaxf(v1, 0.f); }
    out[(size_t)r * NOUT + colA] = v0;
    out[(size_t)r * NOUT + colB] = v1;
  }
}

extern "C" void kernel_launch(void* const* d_in, const int* in_sizes, int n_in,
                              void* d_out, int out_size, void* d_ws,
                              size_t ws_size, hipStream_t stream) {
  (void)in_sizes; (void)n_in; (void)out_size; (void)ws_size;
  const float* x_item = (const float*)d_in[0];   // [100000,128]
  const float* x_user = (const float*)d_in[1];   // [50000,128]
  const int* ii_src = (const int*)d_in[2];
  const int* ii_dst = (const int*)d_in[3];
  const int* iu_src = (const int*)d_in[4];
  const int* iu_dst = (const int*)d_in[5];
  const float* w_l1 = (const float*)d_in[6];
  const float* b1   = (const float*)d_in[7];
  const float* w_r1 = (const float*)d_in[8];
  const float* w_l2 = (const float*)d_in[9];
  const float* b2   = (const float*)d_in[10];
  const float* w_r2 = (const float*)d_in[11];
  const float* w_l3 = (const float*)d_in[12];
  const float* b3   = (const float*)d_in[13];
  const float* w_r3 = (const float*)d_in[14];
  const float* w_lin = (const float*)d_in[15];
  const float* b_lin = (const float*)d_in[16];
  float* out = (float*)d_out;

  constexpr int NI = 100000, NU = 50000, E = 800000, D = 128, H = 256;

  float* ws = (float*)d_ws;
  float* cnt_ii = ws;                       // 100352 floats
  float* cnt_iu = ws + 100352;              // 50176 floats
  float* bufB = ws + 150528;                // 12.8M floats: agg_ii/agg_iu/agg3
  float* bufC = bufB + 12800000;            // 25.6M floats: item_x -> user_x3
  float* bufD = bufC + 25600000;            // 12.8M floats: user_x2

  // conv1: item->item
  zero_f32<<<2048, 256, 0, stream>>>(ws, 150528L + 12800000L);  // counts + bufB
  count_edges<<<(E + 255) / 256, 256, 0, stream>>>(ii_dst, cnt_ii, E);
  count_edges<<<(E + 255) / 256, 256, 0, stream>>>(iu_dst, cnt_iu, E);
  scatter_add<<<E, 128, 0, stream>>>(x_item, ii_src, ii_dst, bufB, D);
  sage_gemm<128, 128, 256, true, true><<<NI / 16, 256, 0, stream>>>(
      bufB, cnt_ii, x_item, w_l1, w_r1, b1, bufC);  // item_x

  // conv2: item->user
  zero_f32<<<2048, 256, 0, stream>>>(bufB, 12800000L);
  scatter_add<<<E, 128, 0, stream>>>(x_item, iu_src, iu_dst, bufB, D);
  sage_gemm<128, 128, 256, true, true><<<NU / 16, 256, 0, stream>>>(
      bufB, cnt_iu, x_user, w_l2, w_r2, b2, bufD);  // user_x2

  // conv3: item_x -> user (same edges/counts as conv2)
  zero_f32<<<2048, 256, 0, stream>>>(bufB, 12800000L);
  scatter_add<<<E, 256, 0, stream>>>(bufC, iu_src, iu_dst, bufB, H);
  sage_gemm<256, 256, 256, true, true><<<NU / 16, 256, 0, stream>>>(
      bufB, cnt_iu, bufD, w_l3, w_r3, b3, bufC);  // user_x3 overwrites item_x

  // final linear HID -> OUT (no relu, no mean scaling, single matrix)
  sage_gemm<256, 0, 64, false, false><<<NU / 16, 64, 0, stream>>>(
      bufC, nullptr, nullptr, w_lin, nullptr, b_lin, out);
}